// Net_88648124991289
// MI455X (gfx1250) — compile-verified
//
#include <hip/hip_runtime.h>
#include <hip/hip_bf16.h>
#include <math.h>

#define NN   200000
#define EE   3200000
#define GG   1000
#define FIN  116
#define HH   64
#define KK   15

typedef float v2f __attribute__((ext_vector_type(2)));
typedef float v8f __attribute__((ext_vector_type(8)));

// ---------------------------------------------------------------------------
// Kernel 1: h = x @ W_gcn  via V_WMMA_F32_16X16X4_F32
// block = 128 threads (4 waves). blockIdx.x = 16-row tile; wave w = 16-col tile.
// A (16x4 f32): lanes 0-15 -> K={0,1}, lanes 16-31 -> K={2,3}, M = lane%16.
// B (4x16 f32): VGPR0/1 rows striped across lanes, mirrored halves.
// D (16x16 f32): VGPR j -> M = j + 8*(lane/16), N = lane%16.
// ---------------------------------------------------------------------------
__global__ void gemm_xw_wmma(const float* __restrict__ x,
                             const float* __restrict__ W,
                             float* __restrict__ h) {
  const int wave = threadIdx.x >> 5;   // 0..3  -> column tile
  const int lane = threadIdx.x & 31;
  const int half = lane >> 4;          // 0/1
  const int lr   = lane & 15;
  const int row0 = blockIdx.x * 16;
  const int col0 = wave * 16;

  const float* __restrict__ xrow = x + (size_t)(row0 + lr) * FIN;

  v8f acc = {};
  for (int k0 = 0; k0 < FIN; k0 += 4) {
    const int ka = k0 + 2 * half;
    v2f a, b;
    a.x = xrow[ka];
    a.y = xrow[ka + 1];
    b.x = W[(size_t)ka       * HH + col0 + lr];
    b.y = W[(size_t)(ka + 1) * HH + col0 + lr];
    acc = __builtin_amdgcn_wmma_f32_16x16x4_f32(
        /*neg_a=*/false, a, /*neg_b=*/false, b,
        /*c_mod=*/(short)0, acc, /*reuse_a=*/false, /*reuse_b=*/false);
  }
#pragma unroll
  for (int j = 0; j < 8; ++j) {
    h[(size_t)(row0 + j + 8 * half) * HH + col0 + lr] = acc[j];
  }
}

// ---------------------------------------------------------------------------
// Kernel 2: degree accumulation (deg[dst] += 1)
// ---------------------------------------------------------------------------
__global__ void deg_kernel(const int* __restrict__ dst, float* __restrict__ deg) {
  int e = blockIdx.x * blockDim.x + threadIdx.x;
  if (e < EE) atomicAdd(&deg[dst[e]], 1.0f);
}

// ---------------------------------------------------------------------------
// Kernel 3: dis[i] = 1/sqrt(deg[i] + 1)   (in place)
// ---------------------------------------------------------------------------
__global__ void dis_kernel(float* __restrict__ deg) {
  int i = blockIdx.x * blockDim.x + threadIdx.x;
  if (i < NN) deg[i] = 1.0f / sqrtf(deg[i] + 1.0f);
}

// ---------------------------------------------------------------------------
// Kernel 4: agg[dst] += h[src] * dis[src]*dis[dst]
// One wave per edge; each lane handles 2 contiguous features (float2 load).
// h and agg are L2-resident (51 MB each << 192 MB L2).
// ---------------------------------------------------------------------------
__global__ void edge_agg(const int* __restrict__ src, const int* __restrict__ dst,
                         const float* __restrict__ h, const float* __restrict__ dis,
                         float* __restrict__ agg) {
  int gid  = blockIdx.x * blockDim.x + threadIdx.x;
  int e    = gid >> 5;
  int lane = threadIdx.x & 31;
  if (e >= EE) return;
  int s = src[e], d = dst[e];
  float nrm = dis[s] * dis[d];
  const float2* hp = (const float2*)(h + (size_t)s * HH);
  float2 v = hp[lane];
  atomicAdd(&agg[(size_t)d * HH + 2 * lane],     v.x * nrm);
  atomicAdd(&agg[(size_t)d * HH + 2 * lane + 1], v.y * nrm);
}

// ---------------------------------------------------------------------------
// Kernel 5: h1 = relu(agg + h*dis^2 + b); pool: sums[batch[i]] += h1, cnt += 1
// h1 never materialized (only consumed by pooling).
// ---------------------------------------------------------------------------
__global__ void finalize_pool(const float* __restrict__ h, const float* __restrict__ agg,
                              const float* __restrict__ dis, const float* __restrict__ b,
                              const int* __restrict__ batch,
                              float* __restrict__ sums, float* __restrict__ cnt) {
  int gid = blockIdx.x * blockDim.x + threadIdx.x;
  if (gid >= NN * HH) return;
  int i = gid / HH;
  int f = gid - i * HH;
  float di = dis[i];
  float v = agg[gid] + h[gid] * di * di + b[f];
  v = fmaxf(v, 0.0f);
  int g = batch[i];
  atomicAdd(&sums[(size_t)g * HH + f], v);
  if (f == 0) atomicAdd(&cnt[g], 1.0f);
}

// ---------------------------------------------------------------------------
// Kernel 6: sub_e = sums / max(cnt, 1)
// ---------------------------------------------------------------------------
__global__ void mean_kernel(const float* __restrict__ sums, const float* __restrict__ cnt,
                            float* __restrict__ sub_e) {
  int gid = blockIdx.x * blockDim.x + threadIdx.x;
  if (gid >= GG * HH) return;
  sub_e[gid] = sums[gid] / fmaxf(cnt[gid / HH], 1.0f);
}

// ---------------------------------------------------------------------------
// Kernel 7: scores[g] = dot(tanh(sub_e[g] @ att_W), h_t)   (one wave / graph)
// ---------------------------------------------------------------------------
__global__ void attn_scores(const float* __restrict__ sub_e, const float* __restrict__ att_W,
                            const float* __restrict__ h_t, float* __restrict__ scores) {
  int gid  = blockIdx.x * blockDim.x + threadIdx.x;
  int g    = gid >> 5;
  int lane = threadIdx.x & 31;
  if (g >= GG) return;
  const float* se = sub_e + (size_t)g * HH;
  float acc = 0.0f;
#pragma unroll
  for (int jj = 0; jj < 2; ++jj) {
    int j = lane + 32 * jj;
    float t = 0.0f;
    for (int k = 0; k < HH; ++k) t += se[k] * att_W[(size_t)k * HH + j];
    acc += tanhf(t) * h_t[j];
  }
  for (int off = 16; off > 0; off >>= 1) acc += __shfl_down(acc, off, 32);
  if (lane == 0) scores[g] = acc;
}

// ---------------------------------------------------------------------------
// Kernel 8: single-block tail — softmax, h_s, y_s, top-15, gathers, h_sub, out
// Output: y_s(2) | att(1000) | a_i(15) | ss_bb(15*116) | h_sub(64) | out(2)
// ---------------------------------------------------------------------------
__global__ void tail_kernel(const float* __restrict__ scores, const float* __restrict__ sub_e,
                            const float* __restrict__ sub, const float* __restrict__ lin1_W,
                            const float* __restrict__ lin1_b, float* __restrict__ out) {
  __shared__ float s_att[GG];
  __shared__ float s_tmp[256];
  __shared__ int   s_idx[KK];
  __shared__ float s_hs[HH];
  __shared__ float s_hsub[HH];
  const int tid = threadIdx.x;

  // max over scores
  float m = -INFINITY;
  for (int g = tid; g < GG; g += 256) m = fmaxf(m, scores[g]);
  s_tmp[tid] = m; __syncthreads();
  for (int s = 128; s > 0; s >>= 1) {
    if (tid < s) s_tmp[tid] = fmaxf(s_tmp[tid], s_tmp[tid + s]);
    __syncthreads();
  }
  float mx = s_tmp[0]; __syncthreads();

  // exp + sum
  float sum = 0.0f;
  for (int g = tid; g < GG; g += 256) {
    float e = expf(scores[g] - mx);
    s_att[g] = e;
    sum += e;
  }
  s_tmp[tid] = sum; __syncthreads();
  for (int s = 128; s > 0; s >>= 1) {
    if (tid < s) s_tmp[tid] += s_tmp[tid + s];
    __syncthreads();
  }
  float tot = s_tmp[0]; __syncthreads();

  for (int g = tid; g < GG; g += 256) {
    s_att[g] /= tot;
    out[2 + g] = s_att[g];
  }
  __syncthreads();

  // h_s = sum_g att[g] * sub_e[g]
  if (tid < HH) {
    float hs = 0.0f;
    for (int g = 0; g < GG; ++g) hs += s_att[g] * sub_e[(size_t)g * HH + tid];
    s_hs[tid] = hs;
  }
  __syncthreads();
  if (tid < 2) {
    float y = lin1_b[tid];
    for (int k = 0; k < HH; ++k) y += s_hs[k] * lin1_W[k * 2 + tid];
    out[tid] = y;
  }

  // top-K (serial argmax; strict '>' matches lax.top_k lowest-index tie-break)
  if (tid == 0) {
    for (int k = 0; k < KK; ++k) {
      float best = -INFINITY; int bi = 0;
      for (int g = 0; g < GG; ++g)
        if (s_att[g] > best) { best = s_att[g]; bi = g; }
      s_idx[k] = bi;
      s_att[bi] = -INFINITY;
      out[2 + GG + k] = (float)bi;
    }
  }
  __syncthreads();

  // ss_bb = sub[a_i]
  for (int t = tid; t < KK * FIN; t += 256) {
    int k = t / FIN, j = t - k * FIN;
    out[2 + GG + KK + t] = sub[(size_t)s_idx[k] * FIN + j];
  }

  // h_sub = sum_k sub_e[a_i[k]]
  if (tid < HH) {
    float hv = 0.0f;
    for (int k = 0; k < KK; ++k) hv += sub_e[(size_t)s_idx[k] * HH + tid];
    s_hsub[tid] = hv;
    out[2 + GG + KK + KK * FIN + tid] = hv;
  }
  __syncthreads();
  if (tid < 2) {
    float y = lin1_b[tid];
    for (int k = 0; k < HH; ++k) y += s_hsub[k] * lin1_W[k * 2 + tid];
    out[2 + GG + KK + KK * FIN + HH + tid] = y;
  }
}

// ---------------------------------------------------------------------------
extern "C" void kernel_launch(void* const* d_in, const int* in_sizes, int n_in,
                              void* d_out, int out_size, void* d_ws, size_t ws_size,
                              hipStream_t stream) {
  const float* x      = (const float*)d_in[0];
  const int*   eidx   = (const int*)  d_in[1];
  const int*   batch  = (const int*)  d_in[2];
  const float* sub    = (const float*)d_in[3];
  const float* h_t    = (const float*)d_in[4];
  const float* W_gcn  = (const float*)d_in[5];
  const float* b_gcn  = (const float*)d_in[6];
  const float* att_W  = (const float*)d_in[7];
  const float* lin1_W = (const float*)d_in[8];
  const float* lin1_b = (const float*)d_in[9];
  const int* src = eidx;
  const int* dst = eidx + EE;

  char* ws = (char*)d_ws;
  size_t off = 0;
  float* h     = (float*)(ws + off); off += (size_t)NN * HH * 4;   // 51.2 MB
  float* agg   = (float*)(ws + off); off += (size_t)NN * HH * 4;   // 51.2 MB
  float* deg   = (float*)(ws + off); off += (size_t)NN * 4;        // 0.8 MB (becomes dis)
  float* sums  = (float*)(ws + off); off += (size_t)GG * HH * 4;
  float* cnt   = (float*)(ws + off); off += (size_t)GG * 4;
  float* sub_e = (float*)(ws + off); off += (size_t)GG * HH * 4;
  float* scores= (float*)(ws + off); off += (size_t)GG * 4;

  // zero atomic accumulation targets (graph-capturable memset nodes)
  hipMemsetAsync(agg,  0, (size_t)NN * HH * 4, stream);
  hipMemsetAsync(deg,  0, (size_t)NN * 4,      stream);
  hipMemsetAsync(sums, 0, (size_t)GG * HH * 4, stream);
  hipMemsetAsync(cnt,  0, (size_t)GG * 4,      stream);

  // 1) h = x @ W_gcn  (WMMA fp32)
  gemm_xw_wmma<<<NN / 16, 128, 0, stream>>>(x, W_gcn, h);
  // 2) degrees
  deg_kernel<<<(EE + 255) / 256, 256, 0, stream>>>(dst, deg);
  // 3) dis = rsqrt(deg + 1)
  dis_kernel<<<(NN + 255) / 256, 256, 0, stream>>>(deg);
  // 4) edge aggregation (wave per edge)
  edge_agg<<<(EE * 32) / 256, 256, 0, stream>>>(src, dst, h, deg, agg);
  // 5) relu(agg + self-loop + b) fused with mean-pool accumulation
  finalize_pool<<<((size_t)NN * HH + 255) / 256, 256, 0, stream>>>(
      h, agg, deg, b_gcn, batch, sums, cnt);
  // 6) mean
  mean_kernel<<<(GG * HH + 255) / 256, 256, 0, stream>>>(sums, cnt, sub_e);
  // 7) attention scores (wave per graph)
  attn_scores<<<(GG * 32 + 255) / 256, 256, 0, stream>>>(sub_e, att_W, h_t, scores);
  // 8) softmax + heads + top-k tail
  tail_kernel<<<1, 256, 0, stream>>>(scores, sub_e, sub, lin1_W, lin1_b, (float*)d_out);
}